// QuantNorm1D_52424370815663
// MI455X (gfx1250) — compile-verified
//
#include <hip/hip_runtime.h>
#include <hip/hip_bf16.h>
#include <math.h>
#include <stdint.h>

// KDE with per-sample Silverman bandwidth.
// B=128 rows, F=512 features, V=4096 kernel centers (rows 0..127 aliased to x).
// Compute-bound: ~268M erf evals; all data L2-resident (8.5 MB) -> VALU roof,
// not HBM (23.3 TB/s would cover the 8.5 MB in <1us). No matmul structure, so
// WMMA is inapplicable; the CDNA5 feature that pays is the ASYNCcnt pipeline:
// double-buffered global_load_async_to_lds_b128 staging of value tiles.

#define KDE_B 128
#define KDE_F 512
#define KDE_V 4096
#define TPB   256          // threads per block in main kernel (8 wave32)
#define TV    16           // v-rows per tile (128 % 16 == 0 -> no straddle of x/values)
#define NTILES (KDE_V / TV)

#if defined(__has_builtin)
#  if __has_builtin(__builtin_amdgcn_global_load_async_to_lds_b128)
#    define HAVE_ASYNC_LDS 1
#  endif
#  if __has_builtin(__builtin_amdgcn_s_wait_asynccnt)
#    define HAVE_WAIT_ASYNC 1
#  endif
#endif
#ifndef HAVE_ASYNC_LDS
#  define HAVE_ASYNC_LDS 0
#endif
#ifndef HAVE_WAIT_ASYNC
#  define HAVE_WAIT_ASYNC 0
#endif

// The async-LDS builtin takes typed v4i pointers in explicit address spaces:
// (global v4i*, lds v4i*, imm offset, imm cpol).
typedef int v4i __attribute__((ext_vector_type(4)));
typedef __attribute__((address_space(1))) v4i glob_v4i;
typedef __attribute__((address_space(3))) v4i lds_v4i;

// Flat LDS-aperture addresses carry the LDS byte offset in [31:0] (ISA 10.2),
// so truncation gives a valid AS(3) pointer.
__device__ __forceinline__ glob_v4i* to_glob_v4(const void* p) {
  return (glob_v4i*)(uintptr_t)p;
}
__device__ __forceinline__ lds_v4i* to_lds_v4(const void* p) {
  return (lds_v4i*)(uint32_t)(uintptr_t)p;
}

__device__ __forceinline__ void wait_async_le4() {
#if HAVE_ASYNC_LDS
#  if HAVE_WAIT_ASYNC
  __builtin_amdgcn_s_wait_asynccnt(4);
#  else
  asm volatile("s_wait_asynccnt 4" ::: "memory");
#  endif
#endif
}
__device__ __forceinline__ void wait_async_le0() {
#if HAVE_ASYNC_LDS
#  if HAVE_WAIT_ASYNC
  __builtin_amdgcn_s_wait_asynccnt(0);
#  else
  asm volatile("s_wait_asynccnt 0" ::: "memory");
#  endif
#endif
}

// ---------------------------------------------------------------------------
// Main kernel (placed first in .text for disasm visibility):
// out[b,f] = (1/F) * sum_v Phi((x[b,f]-vals[v,f]) / bw[b])
// Grid: (B * F/TPB) blocks; block handles one row b and 256 features.
// v-tiles of 16 rows double-buffered in LDS via async global->LDS loads.
// ---------------------------------------------------------------------------
__global__ __launch_bounds__(TPB) void kde_main_kernel(
    const float* __restrict__ x, const float* __restrict__ values,
    const float* __restrict__ inv_bw, float* __restrict__ out) {
  __shared__ __align__(16) float tile[2][TV * TPB];  // 2 x 16KB

  const int tid = threadIdx.x;
  const int b   = blockIdx.x >> 1;          // F/TPB == 2 tiles per row
  const int f0  = (blockIdx.x & 1) * TPB;
  const int f   = f0 + tid;

  const float xf    = x[b * KDE_F + f];
  // Fold 1/sqrt(2) into the bandwidth scale: erf(z/sqrt2), z=(xf-v)*ibw.
  const float scale = inv_bw[b] * 0.70710678118654752f;

  // Stage v-tile tt (rows tt*TV..tt*TV+15, cols f0..f0+255) into tile[buf].
  // 16KB tile = 4 x b128 per thread (8 waves x 4 instr, 512B per wave-instr).
  auto issue_tile = [&](int tt, int buf) {
    const int v0 = tt * TV;
    const float* src = (v0 < KDE_B) ? (x + (size_t)v0 * KDE_F + f0)
                                    : (values + (size_t)v0 * KDE_F + f0);
#pragma unroll
    for (int i = 0; i < 4; ++i) {
      const int e   = (i * TPB + tid) * 4;   // flat float index in tile [0,4096)
      const int row = e >> 8;                 // / 256 cols
      const int col = e & 255;
#if HAVE_ASYNC_LDS
      __builtin_amdgcn_global_load_async_to_lds_b128(
          to_glob_v4(src + (size_t)row * KDE_F + col),
          to_lds_v4(&tile[buf][e]), 0, 0);
#else
      *(float4*)&tile[buf][e] = *(const float4*)(src + (size_t)row * KDE_F + col);
#endif
    }
  };

  issue_tile(0, 0);

  float acc = 0.0f;   // accumulates raw erf values in [-1,1]
  for (int t = 0; t < NTILES; ++t) {
    const int cur = t & 1;
    if (t + 1 < NTILES) {
      issue_tile(t + 1, cur ^ 1);   // prefetch next tile into other buffer
      wait_async_le4();             // >=4 outstanding belong to next tile;
                                    // in-order completion => current tile done
    } else {
      wait_async_le0();
    }
    __syncthreads();                // all waves' portions of tile[cur] landed

#pragma unroll
    for (int r = 0; r < TV; ++r) {
      acc += erff((xf - tile[cur][r * TPB + tid]) * scale);
    }
    __syncthreads();                // protect tile[cur] before it is re-filled
  }

  // sum_v 0.5*(1+erf) / F  ==  0.5*acc/F + V/(2F)
  out[b * KDE_F + f] = acc * (0.5f / (float)KDE_F)
                     + (float)KDE_V * (0.5f / (float)KDE_F);
}

// ---------------------------------------------------------------------------
// Stats kernel: per-row -> inv_bw[b] = 1 / (0.9*min(std, IQR/1.34)*V^-0.2)
// One 512-thread block (16 wave32) per row. Exact quantiles via bitonic sort.
// ---------------------------------------------------------------------------
__global__ __launch_bounds__(KDE_F) void kde_stats_kernel(
    const float* __restrict__ x, float* __restrict__ inv_bw) {
  __shared__ float s_sort[KDE_F];
  __shared__ float s_red[KDE_F];
  const int t = threadIdx.x;
  const int b = blockIdx.x;

  const float v = x[b * KDE_F + t];
  s_sort[t] = v;
  s_red[t]  = v;
  __syncthreads();

  // mean
  for (int s = KDE_F / 2; s > 0; s >>= 1) {
    if (t < s) s_red[t] += s_red[t + s];
    __syncthreads();
  }
  const float mean = s_red[0] * (1.0f / (float)KDE_F);
  __syncthreads();

  // variance (ddof=1)
  const float d = v - mean;
  s_red[t] = d * d;
  __syncthreads();
  for (int s = KDE_F / 2; s > 0; s >>= 1) {
    if (t < s) s_red[t] += s_red[t + s];
    __syncthreads();
  }
  const float stdv = sqrtf(s_red[0] / (float)(KDE_F - 1));

  // bitonic sort of the row (512 elems, 512 threads)
  for (int k = 2; k <= KDE_F; k <<= 1) {
    for (int j = k >> 1; j > 0; j >>= 1) {
      __syncthreads();
      const int ixj = t ^ j;
      if (ixj > t) {
        const float a = s_sort[t];
        const float c = s_sort[ixj];
        const bool up = ((t & k) == 0);
        if ((a > c) == up) { s_sort[t] = c; s_sort[ixj] = a; }
      }
    }
  }
  __syncthreads();

  if (t == 0) {
    // jnp.quantile 'linear': pos25=127.75, pos75=383.25
    const float q25 = 0.25f * s_sort[127] + 0.75f * s_sort[128];
    const float q75 = 0.75f * s_sort[383] + 0.25f * s_sort[384];
    // 4096^-0.2 = 2^-2.4
    const float bw = 0.9f * fminf(stdv, (q75 - q25) * (1.0f / 1.34f))
                          * 0.18946457081379978f;
    inv_bw[b] = 1.0f / bw;
  }
}

// ---------------------------------------------------------------------------
extern "C" void kernel_launch(void* const* d_in, const int* in_sizes, int n_in,
                              void* d_out, int out_size, void* d_ws, size_t ws_size,
                              hipStream_t stream) {
  (void)in_sizes; (void)n_in; (void)out_size; (void)ws_size;
  const float* x      = (const float*)d_in[0];   // (128, 512) f32
  const float* values = (const float*)d_in[1];   // (4096, 512) f32
  float*       out    = (float*)d_out;           // (128, 512) f32
  float*       inv_bw = (float*)d_ws;            // 128 floats scratch

  kde_stats_kernel<<<KDE_B, KDE_F, 0, stream>>>(x, inv_bw);
  kde_main_kernel<<<KDE_B * (KDE_F / TPB), TPB, 0, stream>>>(x, values, inv_bw, out);
}